// LightGCN_SimGCL_9028021256887
// MI455X (gfx1250) — compile-verified
//
#include <hip/hip_runtime.h>
#include <cstdint>

// LightGCN 3-layer propagation on MI455X (gfx1250, wave32).
// Memory/atomic-bound: ~7 GB traffic -> ~0.3 ms roofline @ 23.3 TB/s.
// No dense matmul structure => no WMMA; this is an async/atomic/memory-path
// kernel:
//   - async-to-LDS double-buffered edge-stream staging (ASYNCcnt pipeline)
//   - global_prefetch_b8 on gather rows before the dependent loop
//   - v_readlane wave32 broadcast -> scalar (SGPR) gather/scatter bases
//   - native fp32 L2 atomics for the scatter-add

typedef float v4f __attribute__((ext_vector_type(4)));

static constexpr int kUsers  = 100000;
static constexpr int kItems  = 50000;
static constexpr int kNodes  = kUsers + kItems;   // 150000
static constexpr int kDim    = 64;
static constexpr int kEdges  = 4000000;
static constexpr int kLayers = 3;

static constexpr int kVec4         = kNodes * kDim / 4;       // 2,400,000 v4f
static constexpr int kChunks       = 5;                       // 32-edge chunks per wave
static constexpr int kWavesPerBlk  = 8;                       // 256 threads
static constexpr int kEdgesPerWave = 32 * kChunks;            // 160
static constexpr int kSpmmBlocks   = kEdges / (kEdgesPerWave * kWavesPerBlk); // 3125 exact

// One async DMA: each of the 32 lanes copies 4B global -> 4B LDS (ASYNCcnt).
__device__ __forceinline__ void async_ld_b32(unsigned lds_byte, const void* gptr) {
    asm volatile("global_load_async_to_lds_b32 %0, %1, off"
                 :: "v"(lds_byte), "v"(gptr)
                 : "memory");
}

// d1[i] = d2[i] = s[i]; the empty asm pins v in registers so the compiler
// cannot rematerialize the load (observed double global_load_b128 otherwise).
__global__ __launch_bounds__(256) void lgcn_copy2(const v4f* __restrict__ s,
                                                  v4f* __restrict__ d1,
                                                  v4f* __restrict__ d2) {
    const int i = blockIdx.x * blockDim.x + threadIdx.x;
    v4f v = s[i];
    asm volatile("" : "+v"(v));
    d1[i] = v;
    d2[i] = v;
}

__global__ __launch_bounds__(256) void lgcn_zero(v4f* __restrict__ p) {
    const int i = blockIdx.x * blockDim.x + threadIdx.x;
    p[i] = (v4f)(0.0f);
}

// One wave32 per 160 edges, processed as 5 chunks of 32 with double-buffered
// async-to-LDS staging of (row, col, w). Per chunk: lane loads its edge triple
// from LDS, v_readlane broadcasts each edge to SGPRs, 32 lanes cover the
// 64-float source row as float2 each (256B coalesced) and scatter with native
// fp32 global atomics.
__global__ __launch_bounds__(256) void lgcn_spmm(const float* __restrict__ emb,
                                                 float* __restrict__ out,
                                                 const int* __restrict__ row,
                                                 const int* __restrict__ col,
                                                 const float* __restrict__ wt) {
    __shared__ int sbuf[kWavesPerBlk * 2 * 96];   // [wave][buf][r[32] c[32] w[32]]
    const int lane = threadIdx.x & 31;
    const int wib  = threadIdx.x >> 5;
    const int wave = (blockIdx.x * blockDim.x + threadIdx.x) >> 5;
    const int base = wave * kEdgesPerWave;        // kEdges divides exactly

    int* mybuf = &sbuf[wib * 192];
    // Flat LDS addresses carry the LDS byte offset in the low 32 bits.
    const unsigned lds0 = (unsigned)(uintptr_t)mybuf + (unsigned)(lane * 4);

    auto issue = [&](int chunk, int b) {
        const int e = base + chunk * 32 + lane;
        const unsigned o = lds0 + (unsigned)(b * 384);
        async_ld_b32(o,       row + e);
        async_ld_b32(o + 128, col + e);
        async_ld_b32(o + 256, wt  + e);
    };

    issue(0, 0);
    for (int ch = 0; ch < kChunks; ++ch) {
        const int b = ch & 1;
        if (ch + 1 < kChunks) {
            issue(ch + 1, b ^ 1);                              // 6 in flight
            asm volatile("s_wait_asynccnt 0x3" ::: "memory");  // chunk ch landed
        } else {
            asm volatile("s_wait_asynccnt 0x0" ::: "memory");
        }

        const int   r_ = mybuf[b * 96 +  0 + lane];
        const int   c_ = mybuf[b * 96 + 32 + lane];
        const float w_ = __int_as_float(mybuf[b * 96 + 64 + lane]);

        // Pull all 32 gather rows of this chunk toward the caches
        // (global_prefetch_b8; each 256B row = 2 cachelines).
        __builtin_prefetch(emb + (size_t)c_ * kDim, 0, 0);
        __builtin_prefetch(emb + (size_t)c_ * kDim + 32, 0, 0);

#pragma unroll
        for (int j = 0; j < 32; ++j) {
            const int   c  = __builtin_amdgcn_readlane(c_, j);
            const int   r  = __builtin_amdgcn_readlane(r_, j);
            const float wj = __uint_as_float(
                (unsigned)__builtin_amdgcn_readlane((int)__float_as_uint(w_), j));

            const float2 m = ((const float2*)(emb + (size_t)c * kDim))[lane];
            float* dst = out + (size_t)r * kDim + lane * 2;
            __hip_atomic_fetch_add(dst + 0, m.x * wj, __ATOMIC_RELAXED, __HIP_MEMORY_SCOPE_AGENT);
            __hip_atomic_fetch_add(dst + 1, m.y * wj, __ATOMIC_RELAXED, __HIP_MEMORY_SCOPE_AGENT);
        }
    }
}

// acc = (acc + layer) * scale, and zero the retired ping-pong buffer so it is
// ready to be the next layer's scatter target (fuses the per-layer zero pass).
__global__ __launch_bounds__(256) void lgcn_add_scale_zero(v4f* __restrict__ acc,
                                                           const v4f* __restrict__ layer,
                                                           v4f* __restrict__ tozero,
                                                           float scale) {
    const int i = blockIdx.x * blockDim.x + threadIdx.x;
    acc[i] = (acc[i] + layer[i]) * scale;
    tozero[i] = (v4f)(0.0f);
}

// Last layer: no buffer to recycle.
__global__ __launch_bounds__(256) void lgcn_add_scale(v4f* __restrict__ acc,
                                                      const v4f* __restrict__ layer,
                                                      float scale) {
    const int i = blockIdx.x * blockDim.x + threadIdx.x;
    acc[i] = (acc[i] + layer[i]) * scale;
}

extern "C" void kernel_launch(void* const* d_in, const int* in_sizes, int n_in,
                              void* d_out, int out_size, void* d_ws, size_t ws_size,
                              hipStream_t stream) {
    const float* user_emb = (const float*)d_in[0];           // [100000*64]
    const float* item_emb = (const float*)d_in[1];           // [50000*64]
    const float* edge_w   = (const float*)d_in[2];           // [4M]
    const int*   edge_idx = (const int*)d_in[3];             // [2, 4M]
    const int*   row      = edge_idx;
    const int*   col      = edge_idx + kEdges;

    float* acc  = (float*)d_out;                             // [150000*64]
    float* bufA = (float*)d_ws;                              // cur
    float* bufB = bufA + (size_t)kNodes * kDim;              // nxt (2 x 38.4 MB in ws)

    const int denseBlocks = kVec4 / 256;                     // 9375, exact
    const int userBlocks  = kUsers * kDim / 4 / 256;         // 6250, exact
    const int itemBlocks  = kItems * kDim / 4 / 256;         // 3125, exact

    lgcn_copy2<<<userBlocks, 256, 0, stream>>>(
        (const v4f*)user_emb, (v4f*)bufA, (v4f*)acc);
    lgcn_copy2<<<itemBlocks, 256, 0, stream>>>(
        (const v4f*)item_emb,
        (v4f*)(bufA + (size_t)kUsers * kDim),
        (v4f*)(acc  + (size_t)kUsers * kDim));
    lgcn_zero<<<denseBlocks, 256, 0, stream>>>((v4f*)bufB);  // only standalone zero

    float* cur = bufA;
    float* nxt = bufB;
    for (int l = 0; l < kLayers; ++l) {
        lgcn_spmm<<<kSpmmBlocks, 256, 0, stream>>>(cur, nxt, row, col, edge_w);
        if (l == kLayers - 1) {
            lgcn_add_scale<<<denseBlocks, 256, 0, stream>>>(
                (v4f*)acc, (const v4f*)nxt, 1.0f / (kLayers + 1));
        } else {
            // recycle `cur` as the next layer's zeroed scatter target
            lgcn_add_scale_zero<<<denseBlocks, 256, 0, stream>>>(
                (v4f*)acc, (const v4f*)nxt, (v4f*)cur, 1.0f);
        }
        float* t = cur; cur = nxt; nxt = t;
    }
}